// HieRecClickPredictor_42949673192
// MI455X (gfx1250) — compile-verified
//
#include <hip/hip_runtime.h>

// HieRec click predictor, MI455X (gfx1250, wave32).
// out[i] = dot(V[i,:], 0.7*sw*Sub[s,:] + 0.15*cw*Cat[c,:] + 0.15*User[b,:])
// Memory-bound (~500 MB HBM -> ~21 us @ 23.3 TB/s). The per-item dot products
// are computed 16-at-a-time as the diagonal of V * Comb^T via
// v_wmma_f32_16x16x4_f32 (extra off-diagonal FLOPs are free vs. the
// bandwidth bound). f32 WMMA lane layout (ISA 7.12.2):
//   A 16x4 : lane L<16 -> {K=d,d+1} of row M=L ; lane L+16 -> {K=d+2,d+3}
//   B 4x16 : lane L<16 -> {K=d,d+1} of col N=L ; lane L+16 -> {K=d+2,d+3}
// -> identical per-lane addressing for A and B; every element read once.

typedef __attribute__((ext_vector_type(2))) float v2f;
typedef __attribute__((ext_vector_type(8))) float v8f;

#define NB   4096
#define NN   50
#define ND   256
#define NS   300
#define NC   18
#define LAM_S 0.7f
#define LAM_T 0.15f

__global__ __launch_bounds__(256) void hierec_wmma_kernel(
    const float* __restrict__ vectors,      // [B,N,D]
    const int*   __restrict__ subcategory,  // [B,N]
    const int*   __restrict__ category,     // [B,N]
    const float* __restrict__ sub_repr,     // [B,S,D]
    const float* __restrict__ sub_weights,  // [B,S]
    const float* __restrict__ cat_repr,     // [B,C,D]
    const float* __restrict__ cat_weights,  // [B,C]
    const float* __restrict__ user_repr,    // [B,D]
    float*       __restrict__ out)          // [B,N]
{
    __shared__ float diag_lds[8][16];

    const int lane = threadIdx.x & 31;
    const int wave = threadIdx.x >> 5;
    const int L    = lane & 15;     // item within the wave's 16-item tile
    const int half = lane >> 4;     // K-pair select: 0 -> {d,d+1}, 1 -> {d+2,d+3}

    const int itemBase = (blockIdx.x * 8 + wave) * 16;  // B*N % 128 == 0
    const int i = itemBase + L;                         // flat (b,n)
    const int b = i / NN;

    const int s = subcategory[i];
    const int c = category[i];
    const float a0 = LAM_S * sub_weights[b * NS + s];
    const float a1 = LAM_T * cat_weights[b * NC + c];
    const float a2 = 1.0f - LAM_S - LAM_T;

    const float* pV = vectors  + (size_t)i * ND                    + 2 * half;
    const float* pS = sub_repr + ((size_t)b * NS + s) * (size_t)ND + 2 * half;
    const float* pC = cat_repr + ((size_t)b * NC + c) * (size_t)ND + 2 * half;
    const float* pU = user_repr + (size_t)b * ND                   + 2 * half;

    v8f acc = {};   // 16x16 f32 accumulator (8 VGPRs)

#pragma unroll 4
    for (int d = 0; d < ND; d += 4) {
        v2f v  = *(const v2f*)(pV + d);
        v2f sv = *(const v2f*)(pS + d);
        v2f cv = *(const v2f*)(pC + d);
        v2f uv = *(const v2f*)(pU + d);
        v2f comb;
        comb.x = __builtin_fmaf(a0, sv.x, __builtin_fmaf(a1, cv.x, a2 * uv.x));
        comb.y = __builtin_fmaf(a0, sv.y, __builtin_fmaf(a1, cv.y, a2 * uv.y));
        // D(16x16,f32) += A(16x4,f32) x B(4x16,f32)
        acc = __builtin_amdgcn_wmma_f32_16x16x4_f32(
            /*neg_a=*/false, v, /*neg_b=*/false, comb,
            /*c_mod=*/(short)0, acc, /*reuse_a=*/false, /*reuse_b=*/false);
    }

    // Diagonal of C/D (ISA 7.12.2): element (m,m) lives at
    //   m < 8 : VGPR m,   lane m
    //   m >= 8: VGPR m-8, lane m+16
#pragma unroll
    for (int r = 0; r < 8; ++r) {
        if (lane == r)      diag_lds[wave][r]     = acc[r];
        if (lane == r + 24) diag_lds[wave][r + 8] = acc[r];
    }
    __syncthreads();

    if (half == 0)
        out[itemBase + L] = diag_lds[wave][L];   // coalesced 64B store per wave
}

extern "C" void kernel_launch(void* const* d_in, const int* in_sizes, int n_in,
                              void* d_out, int out_size, void* d_ws, size_t ws_size,
                              hipStream_t stream)
{
    const float* vectors     = (const float*)d_in[0];
    const int*   subcategory = (const int*)  d_in[1];
    const int*   category    = (const int*)  d_in[2];
    const float* sub_repr    = (const float*)d_in[3];
    const float* sub_weights = (const float*)d_in[4];
    const float* cat_repr    = (const float*)d_in[5];
    const float* cat_weights = (const float*)d_in[6];
    const float* user_repr   = (const float*)d_in[7];
    float* out = (float*)d_out;

    // 204800 items / (8 waves * 16 items) = 1600 blocks, no tail.
    dim3 grid((NB * NN) / 128), block(256);
    hierec_wmma_kernel<<<grid, block, 0, stream>>>(
        vectors, subcategory, category, sub_repr, sub_weights,
        cat_repr, cat_weights, user_repr, out);
}